// ConvAE_45131516346722
// MI455X (gfx1250) — compile-verified
//
#include <hip/hip_runtime.h>

typedef __attribute__((ext_vector_type(2))) float v2f;
typedef __attribute__((ext_vector_type(8))) float v8f;

#define B_      8
#define P_      4096
#define D_      256
#define C_      1024
#define PCHUNKS 4
#define PCHUNK  (P_ / PCHUNKS)   // 1024 rows per chunk
#define CT      16               // channels per workgroup (1 wave)
#define LDSP    260              // pitch in floats: 260 % 64 == 4 -> conflict-free b64 frag reads

// ---------------------------------------------------------------------------
// Kernel 1: zero the 32MB output with float4 stores.
// ---------------------------------------------------------------------------
__global__ __launch_bounds__(256) void zero_out_kernel(float4* __restrict__ out, int n4) {
    int i = blockIdx.x * blockDim.x + threadIdx.x;
    if (i < n4) out[i] = make_float4(0.f, 0.f, 0.f, 0.f);
}

// ---------------------------------------------------------------------------
// Kernel 2: fused fp32 WMMA GEMM (scores = x @ W^T) + per-channel running
// argmax. One wave per workgroup; x tiles streamed into a double-buffered LDS
// stage with GLOBAL_LOAD_ASYNC_TO_LDS_B128 (ASYNCcnt DMA), overlapping a
// 4-way-interleaved v_wmma_f32_16x16x4_f32 pipeline on the other buffer.
// ---------------------------------------------------------------------------
__global__ __launch_bounds__(32) void gemm_argmax_kernel(
    const float* __restrict__ x, const float* __restrict__ W,
    float* __restrict__ pval, int* __restrict__ pidx) {

    __shared__ float sW[CT * LDSP];      // 16 channels x 256 (padded)   16.6 KB
    __shared__ float sX[2][16 * LDSP];   // 2 x (16 p-rows x 256 padded) 33.3 KB

    const int pc     = blockIdx.x;            // p-chunk 0..3
    const int cbase  = blockIdx.y * CT;       // channel base
    const int b      = blockIdx.z;            // batch
    const int lane   = threadIdx.x;           // 0..31 (wave32)
    const int pstart = pc * PCHUNK;
    const int pend   = pstart + PCHUNK;

    // Async staging geometry: each lane owns half of one 1KB row (512B),
    // covered by 32 b128 DMA ops; INST_OFFSET advances global+LDS together.
    const int srow  = lane >> 1;              // 0..15
    const int shalf = (lane & 1) * 128;       // float offset in row: 0 / 128

    // --- stage W strip [16 x 256] asynchronously (read only after first wait)
    {
        const unsigned lw = (unsigned)(uintptr_t)&sW[srow * LDSP + shalf];
        const float*   gw = W + (size_t)(cbase + srow) * D_ + shalf;
#pragma unroll
        for (int it = 0; it < 32; ++it)
            asm volatile("global_load_async_to_lds_b128 %0, %1, off offset:%2"
                         :: "v"(lw), "v"(gw), "i"(it * 16) : "memory");
    }

    // --- issue one x tile [16 x 256] into sX[bufi]
    auto issue_tile = [&](int bufi, int p0) {
        const unsigned lx = (unsigned)(uintptr_t)&sX[bufi][srow * LDSP + shalf];
        const float*   gx = x + (size_t)(b * P_ + p0 + srow) * D_ + shalf;
#pragma unroll
        for (int it = 0; it < 32; ++it)
            asm volatile("global_load_async_to_lds_b128 %0, %1, off offset:%2"
                         :: "v"(lx), "v"(gx), "i"(it * 16) : "memory");
    };

    issue_tile(0, pstart);   // prologue: tile 0 in flight behind W

    // A/B fragment addressing (16x4 f32 A, 4x16 f32 B):
    // lane l -> row/col = l%16, K = {0,1} (lanes 0-15) or {2,3} (lanes 16-31).
    const int rowm = lane & 15;
    const int koff = (lane >> 4) * 2;
    const float* bBase = &sW[rowm * LDSP + koff];

    float best  = 0.0f;   // relu floor: all-nonpositive columns resolve to idx 0
    int   bestp = 0;
    int   cur   = 0;

    for (int p0 = pstart; p0 < pend; p0 += 16) {
        // WAR guard: previous tile's LDS reads fully retired before the DMA
        // refills that buffer (dscnt==0 by construction; this is a no-op wait).
        asm volatile("s_wait_dscnt 0" ::: "memory");

        // Keep the pipeline uniform: last iteration issues a dummy reload so
        // the wait-threshold below always leaves exactly the newest 32 pending.
        const int pn = (p0 + 16 < pend) ? (p0 + 16) : pstart;
        issue_tile(cur ^ 1, pn);

        // ASYNC "done" retires in order: <=32 outstanding => W + tile `cur`
        // are resident; only the just-issued 32 may still be in flight.
        asm volatile("s_wait_asynccnt 32" ::: "memory");

        // 4 independent accumulator chains (16 WMMAs each) so consecutive
        // v_wmma ops don't serialize on the accumulation bypass.
        v8f acc0 = {0.f, 0.f, 0.f, 0.f, 0.f, 0.f, 0.f, 0.f};
        v8f acc1 = acc0, acc2 = acc0, acc3 = acc0;
        const float* aBase = &sX[cur][rowm * LDSP + koff];
#pragma unroll
        for (int k = 0; k < D_; k += 16) {
            v2f a0 = *(const v2f*)(aBase + k);
            v2f b0 = *(const v2f*)(bBase + k);
            v2f a1 = *(const v2f*)(aBase + k + 4);
            v2f b1 = *(const v2f*)(bBase + k + 4);
            v2f a2 = *(const v2f*)(aBase + k + 8);
            v2f b2 = *(const v2f*)(bBase + k + 8);
            v2f a3 = *(const v2f*)(aBase + k + 12);
            v2f b3 = *(const v2f*)(bBase + k + 12);
            acc0 = __builtin_amdgcn_wmma_f32_16x16x4_f32(false, a0, false, b0, (short)0, acc0, false, false);
            acc1 = __builtin_amdgcn_wmma_f32_16x16x4_f32(false, a1, false, b1, (short)0, acc1, false, false);
            acc2 = __builtin_amdgcn_wmma_f32_16x16x4_f32(false, a2, false, b2, (short)0, acc2, false, false);
            acc3 = __builtin_amdgcn_wmma_f32_16x16x4_f32(false, a3, false, b3, (short)0, acc3, false, false);
        }
        v8f accs = (acc0 + acc1) + (acc2 + acc3);

        // Epilogue: C/D layout -> VGPR j, lane l holds (M = j + 8*(l>=16), N = l%16).
        const int prow = p0 + (lane >> 4) * 8;
#pragma unroll
        for (int j = 0; j < 8; ++j) {
            float v = accs[j];
            if (v > best) { best = v; bestp = prow + j; }  // strict > keeps first index
        }
        cur ^= 1;
    }

    // Merge lane l with lane l+16 (same channel, other 8-row half).
    float obest = __shfl_xor(best, 16, 32);
    int   obp   = __shfl_xor(bestp, 16, 32);
    if (obest > best || (obest == best && obp < bestp)) { best = obest; bestp = obp; }

    if (lane < 16) {
        int c = cbase + lane;
        size_t o = (size_t)(b * C_ + c) * PCHUNKS + pc;
        pval[o] = best;
        pidx[o] = bestp;
    }
}

// ---------------------------------------------------------------------------
// Kernel 3: reduce chunk partials per (b,c), then out[b, idx, :] += W[c, :].
// ---------------------------------------------------------------------------
__global__ __launch_bounds__(256) void scatter_kernel(
    const float* __restrict__ W, const float* __restrict__ pval,
    const int* __restrict__ pidx, float* __restrict__ out) {

    const int c = blockIdx.x;
    const int b = blockIdx.y;
    __shared__ int sidx;

    if (threadIdx.x == 0) {
        size_t base = (size_t)(b * C_ + c) * PCHUNKS;
        float bv = pval[base];
        int   bi = pidx[base];
        for (int k = 1; k < PCHUNKS; ++k) {   // ascending p-chunks: strict > => first idx wins
            float v = pval[base + k];
            int   i = pidx[base + k];
            if (v > bv) { bv = v; bi = i; }
        }
        sidx = bi;
    }
    __syncthreads();

    const int t = threadIdx.x;  // 256 threads == D_
    float add = W[(size_t)c * D_ + t];
    float* dst = &out[(size_t)(b * P_ + sidx) * D_ + t];
    __hip_atomic_fetch_add(dst, add, __ATOMIC_RELAXED, __HIP_MEMORY_SCOPE_AGENT);
}

// ---------------------------------------------------------------------------
extern "C" void kernel_launch(void* const* d_in, const int* in_sizes, int n_in,
                              void* d_out, int out_size, void* d_ws, size_t ws_size,
                              hipStream_t stream) {
    const float* x = (const float*)d_in[0];   // [8, 4096, 256]
    const float* W = (const float*)d_in[1];   // [1024, 256]
    float* out = (float*)d_out;               // [8, 4096, 256]

    // Workspace: (val, idx) partials per (b, c, p-chunk) = 2 * 128KB.
    float* pval = (float*)d_ws;
    int*   pidx = (int*)((char*)d_ws + sizeof(float) * B_ * C_ * PCHUNKS);

    const int n4 = (B_ * P_ * D_) / 4;
    zero_out_kernel<<<(n4 + 255) / 256, 256, 0, stream>>>((float4*)out, n4);

    gemm_argmax_kernel<<<dim3(PCHUNKS, C_ / CT, B_), 32, 0, stream>>>(x, W, pval, pidx);

    scatter_kernel<<<dim3(C_, B_), 256, 0, stream>>>(W, pval, pidx, out);
}